// SSITrimmedMAELoss_57183194579106
// MI455X (gfx1250) — compile-verified
//
#include <hip/hip_runtime.h>

#define BB 64
#define HH 384
#define WW 512
#define NPX (HH * WW)          // 196608
#define TRIM_IDX 157286        // int(0.8 * 196608)
#define NB 2048                // 11-bit radix histogram

typedef __attribute__((ext_vector_type(2))) float v2f;
typedef __attribute__((ext_vector_type(8))) float v8f;

// ---------- monotonic float <-> key mapping ----------
__device__ __forceinline__ unsigned fkey(float f) {
  unsigned u = __float_as_uint(f);
  return u ^ ((unsigned)((int)u >> 31) | 0x80000000u);
}
__device__ __forceinline__ float fval(unsigned k) {
  unsigned u = (k & 0x80000000u) ? (k ^ 0x80000000u) : ~k;
  return __uint_as_float(u);
}

// ---------- wave32 sum via V_WMMA_F32_16X16X4_F32 (B = ones => row sums) ----------
__device__ __forceinline__ float wave_sum32(float v) {
  v2f a; a[0] = v;    a[1] = 0.0f;
  v2f b; b[0] = 1.0f; b[1] = 1.0f;
  v8f c = {};
  c = __builtin_amdgcn_wmma_f32_16x16x4_f32(false, a, false, b, (short)0, c,
                                            false, false);
  // lanes 0-15 hold D rows 0-7 (any column, all equal), lanes 16-31 rows 8-15
  float s = c[0] + c[1] + c[2] + c[3] + c[4] + c[5] + c[6] + c[7];
  s += __shfl_xor(s, 16, 32);   // combine halves -> every lane has the total
  return s;
}

__device__ __forceinline__ float block_sum(float v, float* sbuf) {
  float s = wave_sum32(v);
  const int lane = threadIdx.x & 31;
  const int wv = threadIdx.x >> 5;
  const int nw = (blockDim.x + 31) >> 5;
  if (lane == 0) sbuf[wv] = s;
  __syncthreads();
  if (wv == 0) {
    float x = (lane < nw) ? sbuf[lane] : 0.0f;
    x = wave_sum32(x);
    if (lane == 0) sbuf[0] = x;
  }
  __syncthreads();
  float r = sbuf[0];
  __syncthreads();
  return r;
}

// ---------- init ----------
__global__ void zero_kernel(float* w, int n) {
  for (int i = threadIdx.x + blockIdx.x * blockDim.x; i < n;
       i += blockDim.x * gridDim.x)
    w[i] = 0.0f;
}

// ---------- per-sample masked median (3-pass radix select) + MAD scale + trim threshold ----------
__global__ void median_scale_kernel(const float* __restrict__ pred,
                                    const float* __restrict__ targ,
                                    float* __restrict__ shiftP,
                                    float* __restrict__ shiftT,
                                    float* __restrict__ scaleP,
                                    float* __restrict__ scaleT,
                                    float* __restrict__ nValid,
                                    float* __restrict__ thrOut) {
  __shared__ unsigned histP[NB];
  __shared__ unsigned histT[NB];
  __shared__ float sbuf[32];
  __shared__ unsigned s_prefP, s_prefT, s_baseP, s_baseT, s_k, s_n;
  __shared__ float s_shP, s_shT;

  const int b = blockIdx.x;
  const int tid = threadIdx.x;
  const int nt = blockDim.x;
  const float* p = pred + (size_t)b * NPX;
  const float* t = targ + (size_t)b * NPX;
  const float4* p4 = (const float4*)p;
  const float4* t4 = (const float4*)t;
  const int n4 = NPX / 4;

  if (tid == 0) { s_prefP = 0u; s_prefT = 0u; s_baseP = 0u; s_baseT = 0u; }
  __syncthreads();

  unsigned localn = 0;
  int hi = 0;  // bits already selected
  const int widths[3] = {11, 11, 10};

  for (int pass = 0; pass < 3; ++pass) {
    const int width = widths[pass];
    const int shift = 32 - hi - width;
    const unsigned bmask = (1u << width) - 1u;

    for (int i = tid; i < NB; i += nt) { histP[i] = 0u; histT[i] = 0u; }
    __syncthreads();
    const unsigned prefP = s_prefP, prefT = s_prefT;

    for (int i = tid; i < n4; i += nt) {
      float4 tv4 = t4[i];
      float4 pv4 = p4[i];
      float tv[4] = {tv4.x, tv4.y, tv4.z, tv4.w};
      float pv[4] = {pv4.x, pv4.y, pv4.z, pv4.w};
#pragma unroll
      for (int j = 0; j < 4; ++j) {
        const bool m = tv[j] > 0.0f;
        const unsigned kt = m ? fkey(tv[j]) : 0xFFFFFFFFu;
        const unsigned kp = m ? fkey(pv[j]) : 0xFFFFFFFFu;
        if (pass == 0) {
          localn += m ? 1u : 0u;
          atomicAdd(&histP[(kp >> shift) & bmask], 1u);
          atomicAdd(&histT[(kt >> shift) & bmask], 1u);
        } else {
          if ((kp >> (32 - hi)) == prefP) atomicAdd(&histP[(kp >> shift) & bmask], 1u);
          if ((kt >> (32 - hi)) == prefT) atomicAdd(&histT[(kt >> shift) & bmask], 1u);
        }
      }
    }
    __syncthreads();

    float nf = 0.0f;
    if (pass == 0) nf = block_sum((float)localn, sbuf);  // uniform branch: all threads

    if (tid == 0) {
      if (pass == 0) {
        const unsigned n = (unsigned)(nf + 0.5f);
        s_n = n;
        s_k = (n > 0u) ? (n - 1u) / 2u : 0u;  // torch-style lower median
      }
      const unsigned k = s_k;
      const unsigned nb = 1u << width;
      unsigned c = s_baseP;
      for (unsigned i2 = 0; i2 < nb; ++i2) {
        const unsigned h = histP[i2];
        if (k < c + h) { s_prefP = (s_prefP << width) | i2; s_baseP = c; break; }
        c += h;
      }
      c = s_baseT;
      for (unsigned i2 = 0; i2 < nb; ++i2) {
        const unsigned h = histT[i2];
        if (k < c + h) { s_prefT = (s_prefT << width) | i2; s_baseT = c; break; }
        c += h;
      }
    }
    hi += width;
    __syncthreads();
  }

  if (tid == 0) {
    const float shP = fval(s_prefP);
    const float shT = fval(s_prefT);
    s_shP = shP; s_shT = shT;
    shiftP[b] = shP; shiftT[b] = shT;
    nValid[b] = (float)s_n;
  }
  __syncthreads();
  const float shP = s_shP, shT = s_shT;

  // 4th scan: masked mean-absolute-deviation sums
  float aP = 0.0f, aT = 0.0f;
  for (int i = tid; i < n4; i += nt) {
    float4 tv4 = t4[i];
    float4 pv4 = p4[i];
    float tv[4] = {tv4.x, tv4.y, tv4.z, tv4.w};
    float pv[4] = {pv4.x, pv4.y, pv4.z, pv4.w};
#pragma unroll
    for (int j = 0; j < 4; ++j) {
      if (tv[j] > 0.0f) {
        aP += fabsf(pv[j] - shP);
        aT += fabsf(tv[j] - shT);
      }
    }
  }
  const float sumP = block_sum(aP, sbuf);
  const float sumT = block_sum(aT, sbuf);

  if (tid == 0) {
    const float nf = (float)s_n;
    const float scP = sumP / nf;
    const float scT = sumT / nf;
    scaleP[b] = scP; scaleT[b] = scT;
    // trim threshold: res at UNSORTED flat index (faithful to reference quirk)
    unsigned maxIdx = (NPX - s_n) + TRIM_IDX;
    if (maxIdx > NPX - 1) maxIdx = NPX - 1;
    const float tv = t[maxIdx];
    const float pv = p[maxIdx];
    const bool m = tv > 0.0f;
    const float pn = (pv - shP) / scP;
    const float tn = (tv - shT) / scT;
    thrOut[b] = m ? fabsf(pn - tn) : 0.0f;
  }
}

// ---------- trimmed MAE sums ----------
__global__ void tmae_kernel(const float* __restrict__ pred,
                            const float* __restrict__ targ,
                            const float* __restrict__ shiftP,
                            const float* __restrict__ shiftT,
                            const float* __restrict__ scaleP,
                            const float* __restrict__ scaleT,
                            const float* __restrict__ thr,
                            float* __restrict__ accum) {
  __shared__ float sbuf[32];
  const int b = blockIdx.y;
  const float shP = shiftP[b], shT = shiftT[b];
  const float iSP = 1.0f / scaleP[b], iST = 1.0f / scaleT[b];
  const float th = thr[b];
  const float4* p4 = (const float4*)(pred + (size_t)b * NPX);
  const float4* t4 = (const float4*)(targ + (size_t)b * NPX);
  const int n4 = NPX / 4;
  const int stride = gridDim.x * blockDim.x;
  float acc = 0.0f;
  for (int i = blockIdx.x * blockDim.x + threadIdx.x; i < n4; i += stride) {
    __builtin_prefetch(&t4[i + stride], 0, 0);   // global_prefetch_b8
    __builtin_prefetch(&p4[i + stride], 0, 0);
    float4 tv4 = t4[i];
    float4 pv4 = p4[i];
    float tv[4] = {tv4.x, tv4.y, tv4.z, tv4.w};
    float pv[4] = {pv4.x, pv4.y, pv4.z, pv4.w};
#pragma unroll
    for (int j = 0; j < 4; ++j) {
      const bool m = tv[j] > 0.0f;
      const float r =
          m ? fabsf((pv[j] - shP) * iSP - (tv[j] - shT) * iST) : 0.0f;
      acc += (r <= th) ? r : 0.0f;   // res > thr -> 0
    }
  }
  const float s = block_sum(acc, sbuf);
  if (threadIdx.x == 0) atomicAdd(&accum[b], s);
}

// ---------- gradient loss at one scale ----------
__global__ void grad_kernel(const float* __restrict__ pred,
                            const float* __restrict__ targ,
                            const float* __restrict__ shiftP,
                            const float* __restrict__ scaleP, int step,
                            float* __restrict__ glAcc,
                            float* __restrict__ gnAcc) {
  __shared__ float sbuf[32];
  const int b = blockIdx.y;
  const int Hs = HH / step, Ws = WW / step;
  const float shP = shiftP[b];
  const float iSP = 1.0f / scaleP[b];
  const float* p = pred + (size_t)b * NPX;
  const float* t = targ + (size_t)b * NPX;
  const int total = Hs * Ws;
  float acc = 0.0f, cnt = 0.0f;
  for (int j = blockIdx.x * blockDim.x + threadIdx.x; j < total;
       j += gridDim.x * blockDim.x) {
    const int y = j / Ws, x = j - y * Ws;
    const int yy = y * step, xx = x * step;
    const float t00 = t[yy * WW + xx];
    const bool m00 = t00 > 0.0f;
    cnt += m00 ? 1.0f : 0.0f;
    const float pn00 = (p[yy * WW + xx] - shP) * iSP;
    if (x + 1 < Ws) {
      const float t01 = t[yy * WW + xx + step];
      if (m00 && (t01 > 0.0f)) {
        const float pn01 = (p[yy * WW + xx + step] - shP) * iSP;
        acc += fabsf(pn01 - pn00);
      }
    }
    if (y + 1 < Hs) {
      const float t10 = t[(yy + step) * WW + xx];
      if (m00 && (t10 > 0.0f)) {
        const float pn10 = (p[(yy + step) * WW + xx] - shP) * iSP;
        acc += fabsf(pn10 - pn00);
      }
    }
  }
  const float s = block_sum(acc, sbuf);
  const float c = block_sum(cnt, sbuf);
  if (threadIdx.x == 0) {
    atomicAdd(glAcc + b, s);
    atomicAdd(gnAcc + b, c);
  }
}

// ---------- final combine: one wave, WMMA batch-64 reduction ----------
__global__ void finalize_kernel(const float* __restrict__ tmae,
                                const float* __restrict__ nValid,
                                const float* __restrict__ gl,
                                const float* __restrict__ gn,
                                float* __restrict__ out) {
  const int lane = threadIdx.x;  // 32 threads
  const int m = lane & 15;
  const int base = (lane < 16) ? 0 : 2;
  float vals[2];
#pragma unroll
  for (int j = 0; j < 2; ++j) {
    const int b = 4 * m + base + j;   // matches 16x4 f32 A-matrix layout
    const float n = nValid[b];
    float tm = 0.0f;
    if (n > 0.0f) tm = tmae[b] / fmaxf(2.0f * n, 1.0f);
    float g = 0.0f;
#pragma unroll
    for (int s = 0; s < 4; ++s) {
      const float nn = gn[s * BB + b];
      if (nn > 0.0f) g += gl[s * BB + b] / fmaxf(nn, 1.0f);
    }
    vals[j] = tm + 0.5f * g;   // ALPHA = 0.5
  }
  v2f a; a[0] = vals[0]; a[1] = vals[1];
  v2f bv; bv[0] = 1.0f;  bv[1] = 1.0f;
  v8f c = {};
  c = __builtin_amdgcn_wmma_f32_16x16x4_f32(false, a, false, bv, (short)0, c,
                                            false, false);
  float s = c[0] + c[1] + c[2] + c[3] + c[4] + c[5] + c[6] + c[7];
  s += __shfl_xor(s, 16, 32);
  if (lane == 0) out[0] = s * (1.0f / 64.0f);   // mean over batch
}

extern "C" void kernel_launch(void* const* d_in, const int* in_sizes, int n_in,
                              void* d_out, int out_size, void* d_ws,
                              size_t ws_size, hipStream_t stream) {
  (void)in_sizes; (void)n_in; (void)out_size; (void)ws_size;
  const float* pred = (const float*)d_in[0];
  const float* targ = (const float*)d_in[1];
  float* out = (float*)d_out;
  float* w = (float*)d_ws;
  float* shiftP = w + 0;
  float* shiftT = w + 64;
  float* scaleP = w + 128;
  float* scaleT = w + 192;
  float* nValid = w + 256;
  float* thr    = w + 320;
  float* tmae   = w + 384;   // 64
  float* gl     = w + 448;   // 4 * 64
  float* gn     = w + 704;   // 4 * 64

  zero_kernel<<<1, 1024, 0, stream>>>(tmae, 64 + 256 + 256);
  median_scale_kernel<<<BB, 1024, 0, stream>>>(pred, targ, shiftP, shiftT,
                                               scaleP, scaleT, nValid, thr);
  tmae_kernel<<<dim3(64, BB), 256, 0, stream>>>(pred, targ, shiftP, shiftT,
                                                scaleP, scaleT, thr, tmae);
  const int chunks[4] = {64, 16, 4, 1};
  for (int s = 0; s < 4; ++s) {
    grad_kernel<<<dim3(chunks[s], BB), 256, 0, stream>>>(
        pred, targ, shiftP, scaleP, 1 << s, gl + s * BB, gn + s * BB);
  }
  finalize_kernel<<<1, 32, 0, stream>>>(tmae, nValid, gl, gn, out);
}